// Smearing_89532888252815
// MI455X (gfx1250) — compile-verified
//
#include <hip/hip_runtime.h>
#include <hip/hip_bf16.h>

// ---------------------------------------------------------------------------
// Shapes (fixed by the reference): N=512, E=64, D=8, V=64, NB=50
// out = [ v (512*64) | e_new (512*512*64) | x (512*3*8) ]  (float32)
// ---------------------------------------------------------------------------
#define N_PTS   512
#define E_DIM   64
#define D_DIM   8
#define NB      50
#define CUT_UP  5.0f
#define DELTA   (CUT_UP / (NB - 1))              // 5/49
#define INV_DELTA (1.0f / DELTA)
#define COEFF   (-0.5f / (DELTA * DELTA))
#define K2PI5   1.2566370614359172f              // 2*pi/5
#define WIN     13                               // 13-bin Gaussian window (±6 sigma)

typedef float v2f __attribute__((ext_vector_type(2)));
typedef float v8f __attribute__((ext_vector_type(8)));

// D = A(16x4 f32) * B(4x16 f32) + C(16x16 f32), full fp32 precision
#define WMMA_F32(a, b, c) \
    __builtin_amdgcn_wmma_f32_16x16x4_f32(false, (a), false, (b), (short)0, (c), false, false)

// ---------------------------------------------------------------------------
// Prep kernel: pre-swizzle the three weight matrices (padded to 64x64) into
// per-lane WMMA B-fragment order so the main loop does one b64 load per frag.
// Fragment layout guess for f32 4x16 B (mirrors documented 16x4 A layout):
//   v0: lanes 0-15 -> row K=4s+0, lanes 16-31 -> row K=4s+2
//   v1: lanes 0-15 -> row K=4s+1, lanes 16-31 -> row K=4s+3
// Stored linearly: floats[ s*256 + t*64 + lane*2 + vv ]
// ---------------------------------------------------------------------------
__global__ void prep_frags(const float* __restrict__ Wfg,
                           const float* __restrict__ Wfc,
                           const float* __restrict__ We,
                           float* __restrict__ ws) {
    int tid = threadIdx.x;
    for (int g = 0; g < 3; ++g) {
        const float* W = (g == 0) ? Wfg : (g == 1) ? Wfc : We;
        float* dst = ws + g * 4096;
        for (int idx = tid; idx < 4096; idx += blockDim.x) {
            int s    = idx >> 8;
            int rem  = idx & 255;
            int t    = rem >> 6;
            int lane = (rem >> 1) & 31;
            int vv   = rem & 1;
            int row  = 4 * s + ((lane < 16) ? vv : vv + 2);   // K index
            int col  = 16 * t + (lane & 15);                  // N index
            float val;
            if (g == 0)      val = (col < NB) ? W[row * NB + col] : 0.0f; // W_fg (E x NB)
            else if (g == 1) val = (row < NB) ? W[row * E_DIM + col] : 0.0f; // W_fc (NB x E)
            else             val = W[row * E_DIM + col];                  // W_e (E x E)
            dst[idx] = val;
        }
    }
}

// ---------------------------------------------------------------------------
// Main fused kernel: one wave32 per 16-pair tile (j0..j0+15 within row i).
// ---------------------------------------------------------------------------
__global__ __launch_bounds__(256) void smear_main(const float* __restrict__ e,
                                                  const float* __restrict__ x,
                                                  const float* __restrict__ be,
                                                  const float* __restrict__ frags,
                                                  float* __restrict__ out) {
    __shared__ float lds[8][16 * 64];           // per-wave s / xf scratch (4KB each)

    const int lane = threadIdx.x & 31;
    const int wave = threadIdx.x >> 5;
    float* S = lds[wave];

    const int T  = blockIdx.x * 8 + wave;       // tile id, 0..16383
    const int i  = T >> 5;                      // row index (512/16 = 32 tiles per row)
    const int j0 = (T & 31) << 4;               // first pair column of tile
    const int lh = lane & 15;
    const bool hi = lane >= 16;

    // ---- zero the s[m][b] table (16x64 floats) -----------------------------
    {
        float4 z = {0.f, 0.f, 0.f, 0.f};
        #pragma unroll
        for (int q = 0; q < 8; ++q)
            *(float4*)&S[lane * 32 + q * 4] = z;
    }

    // ---- scatter Gaussian smear: lane owns (m = lh, d = hi*4 + 0..3) -------
    {
        const int m = lh;
        const int dbase = hi ? 4 : 0;
        const int jrow = j0 + m;
        float xiA[3][4], xjA[3][4];
        #pragma unroll
        for (int c = 0; c < 3; ++c) {
            float4 a = *(const float4*)&x[(size_t)i * 24 + c * 8 + dbase];
            float4 b = *(const float4*)&x[(size_t)jrow * 24 + c * 8 + dbase];
            xiA[c][0] = a.x; xiA[c][1] = a.y; xiA[c][2] = a.z; xiA[c][3] = a.w;
            xjA[c][0] = b.x; xjA[c][1] = b.y; xjA[c][2] = b.z; xjA[c][3] = b.w;
        }
        #pragma unroll
        for (int dd = 0; dd < 4; ++dd) {
            float dx = xjA[0][dd] - xiA[0][dd];
            float dy = xjA[1][dd] - xiA[1][dd];
            float dz = xjA[2][dd] - xiA[2][dd];
            float sq = dx * dx + dy * dy + dz * dz;
            if (sq > 0.f) {
                float dist = sqrtf(sq);
                if (dist < CUT_UP) {
                    float cut = 0.5f * (1.0f - __cosf(K2PI5 * dist));
                    int b0 = (int)floorf(dist * INV_DELTA) - (WIN / 2);
                    b0 = min(max(b0, 0), NB - WIN);
                    float off = b0 * DELTA;
                    #pragma unroll
                    for (int k = 0; k < WIN; ++k) {
                        float t0 = dist - off;
                        float val = cut * __expf(COEFF * t0 * t0);
                        atomicAdd(&S[m * 64 + b0 + k], val);   // ds_add_f32
                        off += DELTA;
                    }
                }
            }
        }
    }
    asm volatile("s_wait_dscnt 0" ::: "memory");   // cross-lane LDS visibility

    // ---- load e-tile A-fragments (16 K-steps, 2 VGPRs each) ----------------
    // A layout (16x4 f32): lanes 0-15 m=l k={4s,4s+1}; lanes 16-31 m=l-16 k={4s+2,4s+3}
    v2f ea[16];
    {
        const float* erow = e + ((size_t)(i * N_PTS + j0 + lh)) * E_DIM + (hi ? 2 : 0);
        #pragma unroll
        for (int s = 0; s < 16; ++s)
            ea[s] = *(const v2f*)(erow + 4 * s);
    }

    // bias, one per output n-tile
    float ber[4];
    #pragma unroll
    for (int t = 0; t < 4; ++t) ber[t] = be[16 * t + lh];

    const v2f* fg = (const v2f*)(frags)         + lane;   // W_fg fragments
    const v2f* fc = (const v2f*)(frags + 4096)  + lane;   // W_fc fragments
    const v2f* fe = (const v2f*)(frags + 8192)  + lane;   // W_e  fragments

    // ---- GEMM-A: filt = e @ W_fg, then xf = filt * s / 8 back into LDS -----
    #pragma unroll
    for (int t = 0; t < 4; ++t) {
        v8f acc = {0.f, 0.f, 0.f, 0.f, 0.f, 0.f, 0.f, 0.f};
        #pragma unroll
        for (int s = 0; s < 16; ++s)
            acc = WMMA_F32(ea[s], fg[(s * 4 + t) * 32], acc);
        // C layout: vgpr r -> m = r (+8 for hi half), n = 16t + lh
        #pragma unroll
        for (int r = 0; r < 8; ++r) {
            int m = r + (hi ? 8 : 0);
            int b = 16 * t + lh;
            float sv = S[m * 64 + b];
            S[m * 64 + b] = acc[r] * sv * 0.125f;     // mean over D=8; also the
        }                                             // C->A layout bounce
    }
    asm volatile("s_wait_dscnt 0" ::: "memory");

    // ---- xf A-fragments from LDS (one b64 per K-step) ----------------------
    v2f xa[16];
    #pragma unroll
    for (int s = 0; s < 16; ++s)
        xa[s] = *(const v2f*)&S[lh * 64 + 4 * s + (hi ? 2 : 0)];

    // ---- output: e_new = e@W_e + xf@W_fc + b_e -----------------------------
    #pragma unroll
    for (int t = 0; t < 4; ++t) {
        v8f acc = {0.f, 0.f, 0.f, 0.f, 0.f, 0.f, 0.f, 0.f};
        #pragma unroll
        for (int s = 0; s < 16; ++s)
            acc = WMMA_F32(ea[s], fe[(s * 4 + t) * 32], acc);
        #pragma unroll
        for (int s = 0; s < 16; ++s)
            acc = WMMA_F32(xa[s], fc[(s * 4 + t) * 32], acc);
        #pragma unroll
        for (int r = 0; r < 8; ++r) {
            int m = r + (hi ? 8 : 0);
            out[((size_t)(i * N_PTS + j0 + m)) * E_DIM + 16 * t + lh] = acc[r] + ber[t];
        }
    }
}

// ---------------------------------------------------------------------------
extern "C" void kernel_launch(void* const* d_in, const int* in_sizes, int n_in,
                              void* d_out, int out_size, void* d_ws, size_t ws_size,
                              hipStream_t stream) {
    const float* v    = (const float*)d_in[0];
    const float* e    = (const float*)d_in[1];
    const float* x    = (const float*)d_in[2];
    const float* W_fg = (const float*)d_in[3];
    const float* W_fc = (const float*)d_in[4];
    const float* W_e  = (const float*)d_in[5];
    const float* b_e  = (const float*)d_in[6];
    float* out = (float*)d_out;

    const size_t v_elems = (size_t)N_PTS * 64;            // 32768
    const size_t e_elems = (size_t)N_PTS * N_PTS * E_DIM; // 16777216
    const size_t x_elems = (size_t)N_PTS * 3 * D_DIM;     // 12288

    // pass-through outputs (graph-capture safe)
    hipMemcpyAsync(out, v, v_elems * sizeof(float), hipMemcpyDeviceToDevice, stream);
    hipMemcpyAsync(out + v_elems + e_elems, x, x_elems * sizeof(float),
                   hipMemcpyDeviceToDevice, stream);

    float* ws = (float*)d_ws;                             // needs 48KB
    prep_frags<<<1, 256, 0, stream>>>(W_fg, W_fc, W_e, ws);

    // 16384 tiles, 8 waves (tiles) per 256-thread block
    smear_main<<<2048, 256, 0, stream>>>(e, x, b_e, ws, out + v_elems);
}